// EUNN_43825846288577
// MI455X (gfx1250) — compile-verified
//
#include <hip/hip_runtime.h>
#include <math.h>

// Problem geometry (fixed by the reference): b=8192, m=2048, c=64.
#define C_LAYERS 64
#define M_DIM    2048
#define PAIRS    1024          // m/2 complex pairs per row
#define ROWS     8             // batch rows per block
#define TPR      64            // threads per row (each owns 16 contiguous pairs)
#define BT       (ROWS * TPR)  // 512 threads = 16 wave32
#define PPT      (PAIRS / TPR) // 16 pairs per thread

// ---------------- CDNA5 async global->LDS helpers ----------------
// Generic LDS pointer low 32 bits are the raw LDS byte offset (ISA 10.2).
__device__ __forceinline__ unsigned lds_off(const void* p) {
    return (unsigned)(unsigned long long)p;
}

// GLOBAL_LOAD_ASYNC_TO_LDS_B128: vdst = LDS byte address VGPR, vaddr = 64-bit
// global address. Tracked by ASYNCcnt. (CDNA5 ISA)
__device__ __forceinline__ void async_ld_b128(unsigned lds_byte, const void* g) {
    asm volatile("global_load_async_to_lds_b128 %0, %1, off"
                 :: "v"(lds_byte), "v"((unsigned long long)g)
                 : "memory");
}

__device__ __forceinline__ void wait_async0() {
#if __has_builtin(__builtin_amdgcn_s_wait_asynccnt)
    __builtin_amdgcn_s_wait_asynccnt(0);
#else
    asm volatile("s_wait_asynccnt 0" ::: "memory");
#endif
}

// XOR swizzle so contiguous-pair LDS reads (lane stride 16 pairs = 256B)
// spread across all 64 banks. Bijective within each 16-pair block.
__device__ __forceinline__ int SL(int p) { return p ^ ((p >> 4) & 15); }

// ---------------- Kernel 1: coefficient table (2 MB, L2-resident) ----------
// cA[l*1024+k] = coef for even position 2k of layer l:  {cp*ct, sp*ct, -cp*st, -sp*st}
// cB[l*1024+k] = coef for odd  position 2k+1 of layer l: {ct, 0, st, 0}
// phi = angles[2k, l], theta = angles[2k+1, l]; angles is (m=2048, c=64) row-major.
__global__ void eunn_coef(const float* __restrict__ angles,
                          float4* __restrict__ cA,
                          float4* __restrict__ cB) {
    int idx = blockIdx.x * blockDim.x + threadIdx.x;   // over 64*1024
    if (idx >= C_LAYERS * PAIRS) return;
    int l = idx >> 10;
    int k = idx & (PAIRS - 1);
    float phi   = angles[(2 * k) * C_LAYERS + l];
    float theta = angles[(2 * k + 1) * C_LAYERS + l];
    float sp, cp, st, ct;
    sincosf(phi, &sp, &cp);
    sincosf(theta, &st, &ct);
    cA[idx] = make_float4(cp * ct, sp * ct, -cp * st, -sp * st);
    cB[idx] = make_float4(ct, 0.0f, st, 0.0f);
}

// ---------------- Kernel 2: fused 64-layer EUNN ----------------
// Block: 8 rows x 64 threads. Thread (r,c) owns pairs p = 16c..16c+15 of its
// row in registers. Per layer: complex 2x2 pair mix (register local), then
// roll by sh = (l odd ? +1 : -1): only one float2 crosses thread boundaries,
// exchanged via a tiny double-buffered LDS ring (1 barrier / layer).
// Coefficients for layer l+1 are async-DMA'd into the other LDS buffer while
// layer l computes.
__global__ __launch_bounds__(BT)
void eunn_main(const float4* __restrict__ x,
               const float4* __restrict__ cA,
               const float4* __restrict__ cB,
               float4* __restrict__ out) {
    __shared__ float4 sA[2][PAIRS];         // 32 KB
    __shared__ float4 sB[2][PAIRS];         // 32 KB
    __shared__ float2 ex[2][ROWS][TPR];     // 8 KB

    const int t = threadIdx.x;
    const int r = t >> 6;                  // row within block
    const int c = t & (TPR - 1);           // column group within row
    const long long gr = (long long)blockIdx.x * ROWS + r;   // global batch row

    // Load my 16 pairs (32 complex numbers) into registers.
    const float4* xrow = x + gr * PAIRS;
    float4 xd[PPT];
#pragma unroll
    for (int q = 0; q < PPT; ++q) xd[q] = xrow[PPT * c + q];

    // Stage layer 0 coefficients into LDS buffer 0 (async DMA).
#pragma unroll
    for (int j = 0; j < 2; ++j) {
        int p = t + BT * j;                // 512 threads cover 1024 pairs
        async_ld_b128(lds_off(&sA[0][SL(p)]), &cA[p]);
        async_ld_b128(lds_off(&sB[0][SL(p)]), &cB[p]);
    }
    wait_async0();
    __syncthreads();

    for (int l = 0; l < C_LAYERS; ++l) {
        const int buf = l & 1;

        // Kick off async staging of layer l+1 into the other buffer.
        if (l + 1 < C_LAYERS) {
            const float4* gA = cA + (long long)(l + 1) * PAIRS;
            const float4* gB = cB + (long long)(l + 1) * PAIRS;
#pragma unroll
            for (int j = 0; j < 2; ++j) {
                int p = t + BT * j;
                async_ld_b128(lds_off(&sA[buf ^ 1][SL(p)]), &gA[p]);
                async_ld_b128(lds_off(&sB[buf ^ 1][SL(p)]), &gB[p]);
            }
        }

        const bool odd = (l & 1) != 0;     // sh = odd ? +1 : -1
        float2 prevY1 = make_float2(0.f, 0.f);
        float2 firstY0 = make_float2(0.f, 0.f);

#pragma unroll
        for (int q = 0; q < PPT; ++q) {
            const int p = PPT * c + q;
            const float4 cfE = sA[buf][SL(p)];   // even position 2p
            const float4 cfO = sB[buf][SL(p)];   // odd  position 2p+1
            const float4 xv  = xd[q];            // {xr0, xi0, xr1, xi1}
            // y[2p]   : x = (xr0,xi0), partner = (xr1,xi1)
            float y0r = xv.x * cfE.x - xv.y * cfE.y + xv.z * cfE.z - xv.w * cfE.w;
            float y0i = xv.x * cfE.y + xv.y * cfE.x + xv.z * cfE.w + xv.w * cfE.z;
            // y[2p+1] : x = (xr1,xi1), partner = (xr0,xi0)
            float y1r = xv.z * cfO.x - xv.w * cfO.y + xv.x * cfO.z - xv.y * cfO.w;
            float y1i = xv.z * cfO.y + xv.w * cfO.x + xv.x * cfO.w + xv.y * cfO.z;

            if (!odd) {
                // sh=-1: new[q] = { y1(q), y0(q+1) }  (one-pair lag finalize)
                if (q == 0) firstY0 = make_float2(y0r, y0i);
                else        xd[q - 1] = make_float4(prevY1.x, prevY1.y, y0r, y0i);
            } else {
                // sh=+1: new[q] = { y1(q-1), y0(q) }; q==0 evens patched later
                if (q == 0) xd[0] = make_float4(0.f, 0.f, y0r, y0i);
                else        xd[q] = make_float4(prevY1.x, prevY1.y, y0r, y0i);
            }
            prevY1 = make_float2(y1r, y1i);
        }

        // Boundary exchange across the 64-thread ring of this row.
        if (!odd) ex[buf][r][c] = firstY0;   // my y0 of first pair -> left neighbor
        else      ex[buf][r][c] = prevY1;    // my y1 of last pair  -> right neighbor

        wait_async0();      // layer l+1 coef landed (my share)
        __syncthreads();    // ex visible + everyone's coef staged

        if (!odd) {
            float2 v = ex[buf][r][(c + 1) & (TPR - 1)];
            xd[PPT - 1] = make_float4(prevY1.x, prevY1.y, v.x, v.y);
        } else {
            float2 v = ex[buf][r][(c + TPR - 1) & (TPR - 1)];
            xd[0].x = v.x;
            xd[0].y = v.y;
        }
    }

    float4* orow = out + gr * PAIRS;
#pragma unroll
    for (int q = 0; q < PPT; ++q) orow[PPT * c + q] = xd[q];
}

// ---------------- Host launcher ----------------
extern "C" void kernel_launch(void* const* d_in, const int* in_sizes, int n_in,
                              void* d_out, int out_size, void* d_ws, size_t ws_size,
                              hipStream_t stream) {
    (void)in_sizes; (void)n_in; (void)out_size; (void)ws_size;
    const float4* x      = (const float4*)d_in[0];   // (8192, 2048, 2) fp32
    const float*  angles = (const float*)d_in[1];    // (2048, 64) fp32
    float4*       out    = (float4*)d_out;

    // Workspace: 2 MB coefficient table (cA | cB), L2-resident thereafter.
    float4* cA = (float4*)d_ws;
    float4* cB = cA + (size_t)C_LAYERS * PAIRS;

    eunn_coef<<<(C_LAYERS * PAIRS) / 256, 256, 0, stream>>>(angles, cA, cB);
    eunn_main<<<8192 / ROWS, BT, 0, stream>>>(x, cA, cB, out);
}